// FAME_GCN_6244882448962
// MI455X (gfx1250) — compile-verified
//
#include <hip/hip_runtime.h>

// ---------------------------------------------------------------------------
// FAME-GCN fused kernel for MI455X (gfx1250, wave32, WMMA)
//
//   out[:, 0:16]  = (M2 + M2^T) @ (feature@W3) + b3,  M2 = sum_k wb2[k]*A[k]
//   out[:,16:32]  = (M9 + M9^T) @ (feature@W1) + b1,  M9 = sum_k wb[k] *A_t[k]
//
// Bandwidth-bound: 1.2 GB of adjacency streamed twice (row + transpose role).
// All heavy math runs through V_WMMA_F32_16X16X4_F32.
//
// Out-of-range rows (last stripe) are handled by CLAMPING the row index:
// WMMA row m of the A-operand only affects row m of the D tile, and tiles
// rows with grow >= NN are never stored, so clamped (wrong) values are
// harmless. This keeps the inner loops branch-free (no exec masking).
// ---------------------------------------------------------------------------

#define NN      5000
#define NFEATC  128
#define NOUT    16
#define NPAD    5008          // 313 * 16
#define NTILES  313           // ceil(5000/16)
#define NWAVES  16            // waves per SpMM block (512 threads)

typedef __attribute__((ext_vector_type(2))) float v2f;
typedef __attribute__((ext_vector_type(8))) float v8f;

// ---------------------------------------------------------------------------
// Kernel 1: S3 = feature @ W3, S1 = feature @ W1   (NPAD rows; pad rows hold
// clamped-row garbage but are never read downstream: B reads rows <= 4999).
// One wave per 16-row stripe. A-operand: lanes 0-15 K={0,1}, 16-31 K={2,3}.
// ---------------------------------------------------------------------------
__global__ __launch_bounds__(32) void feat_gemm_kernel(
    const float* __restrict__ feat,
    const float* __restrict__ W3, const float* __restrict__ W1,
    float* __restrict__ S3, float* __restrict__ S1)
{
    const int lane = threadIdx.x & 31;
    const int I0   = blockIdx.x * 16;
    const int m    = lane & 15;            // M index (and N col index for B/C)
    const int half = (lane >> 4) & 1;
    const int sel  = half << 1;            // K offset: 0 or 2
    const int crow = min(I0 + m, NN - 1);  // clamped: harmless for dead rows
    const float* frow = feat + (size_t)crow * NFEATC;

    v8f acc3 = {}; v8f acc1 = {};
    #pragma unroll
    for (int kb = 0; kb < NFEATC; kb += 4) {
        v2f a = *(const v2f*)(frow + kb + sel);                 // 8B aligned
        v2f b3, b1;
        b3.x = W3[(kb + sel)     * NOUT + m];
        b3.y = W3[(kb + sel + 1) * NOUT + m];
        b1.x = W1[(kb + sel)     * NOUT + m];
        b1.y = W1[(kb + sel + 1) * NOUT + m];
        acc3 = __builtin_amdgcn_wmma_f32_16x16x4_f32(false, a, false, b3,
                                                     (short)0, acc3, false, false);
        acc1 = __builtin_amdgcn_wmma_f32_16x16x4_f32(false, a, false, b1,
                                                     (short)0, acc1, false, false);
    }
    // C layout: VGPR v holds row v (lanes 0-15) / row v+8 (lanes 16-31), col = m.
    #pragma unroll
    for (int v = 0; v < 8; ++v) {
        const int r = I0 + v + 8 * half;   // < NPAD always
        S3[r * NOUT + m] = acc3[v];
        S1[r * NOUT + m] = acc1[v];
    }
}

// ---------------------------------------------------------------------------
// Kernel 2: fused merge + symmetrize + SpMM for one 16-row output stripe.
//   acc = sum_k w_k * ( A_k[I,:] @ S  +  A_k[:,I]^T @ S )
// 512 threads = 16 waves; each wave strides K-chunks of 4; LDS reduction.
// ---------------------------------------------------------------------------
template <int NMAT>
__global__ __launch_bounds__(512) void fused_spmm_kernel(
    const float* __restrict__ adj,     // [NMAT, NN, NN]
    const float* __restrict__ wts,     // [NMAT]
    const float* __restrict__ S,       // [NPAD, 16]
    const float* __restrict__ bias,    // [16]
    float* __restrict__ out,           // [NN, 32]
    const int outOff)
{
    __shared__ float red[NWAVES * 256];

    const int tid  = threadIdx.x;
    const int lane = tid & 31;
    const int wave = tid >> 5;
    const int I0   = blockIdx.x * 16;
    const int m    = lane & 15;
    const int half = (lane >> 4) & 1;
    const int sel  = half << 1;            // 0 or 2
    const int crow = min(I0 + m, NN - 1);  // clamped: dead tile rows only

    float w[NMAT];
    #pragma unroll
    for (int k = 0; k < NMAT; ++k) w[k] = wts[k];

    v8f acc = {};

    // ---- row contribution: sum_k w_k * A_k[I, :] @ S ----
    {
        const float* arow = adj + (size_t)crow * NN + sel;
        #pragma unroll 2
        for (int jb = wave * 4; jb < NN; jb += NWAVES * 4) {
            v2f msum = {0.f, 0.f};
            #pragma unroll
            for (int k = 0; k < NMAT; ++k) {
                const float* p = arow + (size_t)k * ((size_t)NN * NN) + jb;
                __builtin_prefetch(p + NWAVES * 4, 0, 1);
                const v2f a = *(const v2f*)p;                 // 8B aligned
                msum.x = fmaf(w[k], a.x, msum.x);
                msum.y = fmaf(w[k], a.y, msum.y);
            }
            v2f b;
            b.x = S[(jb + sel)     * NOUT + m];
            b.y = S[(jb + sel + 1) * NOUT + m];
            acc = __builtin_amdgcn_wmma_f32_16x16x4_f32(false, msum, false, b,
                                                        (short)0, acc, false, false);
        }
    }

    // ---- column (transpose) contribution: sum_k w_k * A_k[:, I]^T @ S ----
    {
        #pragma unroll 2
        for (int jb = wave * 4; jb < NN; jb += NWAVES * 4) {
            v2f msum = {0.f, 0.f};
            #pragma unroll
            for (int k = 0; k < NMAT; ++k) {
                const float* p = adj + (size_t)k * ((size_t)NN * NN)
                                     + (size_t)(jb + sel) * NN + crow;
                v2f a;
                a.x = p[0];            // coalesced 64B per half-wave
                a.y = p[NN];
                msum.x = fmaf(w[k], a.x, msum.x);
                msum.y = fmaf(w[k], a.y, msum.y);
            }
            v2f b;
            b.x = S[(jb + sel)     * NOUT + m];
            b.y = S[(jb + sel + 1) * NOUT + m];
            acc = __builtin_amdgcn_wmma_f32_16x16x4_f32(false, msum, false, b,
                                                        (short)0, acc, false, false);
        }
    }

    // ---- cross-wave reduction via LDS, then bias + store ----
    #pragma unroll
    for (int v = 0; v < 8; ++v)
        red[wave * 256 + v * 32 + lane] = acc[v];
    __syncthreads();

    if (tid < 256) {
        float sum = 0.f;
        #pragma unroll
        for (int wv = 0; wv < NWAVES; ++wv) sum += red[wv * 256 + tid];
        const int v = tid >> 5;
        const int l = tid & 31;
        const int r = v + 8 * ((l >> 4) & 1);  // C layout row
        const int c = l & 15;                  // C layout col
        const int grow = I0 + r;
        if (grow < NN)
            out[(size_t)grow * 32 + outOff + c] = sum + bias[c];
    }
}

// ---------------------------------------------------------------------------
// Host launch
// ---------------------------------------------------------------------------
extern "C" void kernel_launch(void* const* d_in, const int* in_sizes, int n_in,
                              void* d_out, int out_size, void* d_ws, size_t ws_size,
                              hipStream_t stream) {
    (void)in_sizes; (void)n_in; (void)out_size; (void)ws_size;

    const float* feature = (const float*)d_in[0];   // [5000,128]
    const float* A       = (const float*)d_in[1];   // [3,5000,5000]
    const float* A_t     = (const float*)d_in[2];   // [9,5000,5000]
    const float* wb2     = (const float*)d_in[3];   // [3]
    const float* wb      = (const float*)d_in[4];   // [9]
    const float* W3      = (const float*)d_in[5];   // [128,16]
    const float* b3      = (const float*)d_in[6];   // [16]
    const float* W1      = (const float*)d_in[7];   // [128,16]
    const float* b1      = (const float*)d_in[8];   // [16]
    float*       out     = (float*)d_out;           // [5000,32]

    float* S3 = (float*)d_ws;                       // [NPAD,16]
    float* S1 = S3 + (size_t)NPAD * NOUT;           // [NPAD,16]

    feat_gemm_kernel<<<NTILES, 32, 0, stream>>>(feature, W3, W1, S3, S1);
    fused_spmm_kernel<3><<<NTILES, 512, 0, stream>>>(A,   wb2, S3, b3, out, 0);
    fused_spmm_kernel<9><<<NTILES, 512, 0, stream>>>(A_t, wb,  S1, b1, out, 16);
}